// GIN_51187420233783
// MI455X (gfx1250) — compile-verified
//
#include <hip/hip_runtime.h>

#define D 128
#define TSTRIDE 132   // padded LDS row stride (floats) to avoid bank conflicts

typedef float v2f __attribute__((ext_vector_type(2)));
typedef float v8f __attribute__((ext_vector_type(8)));

// ---------------------------------------------------------------------------
// Aggregation: one wave (32 lanes) per edge; each lane moves 4 floats of the
// 128-float source row and scatter-adds them into agg[dst].  Rows are 512B,
// fully coalesced; the whole h/agg working set (~102MB) is L2-resident.
// ---------------------------------------------------------------------------
__global__ __launch_bounds__(256)
void gin_scatter_kernel(const float* __restrict__ h,
                        const int*   __restrict__ src,
                        const int*   __restrict__ dst,
                        float*       __restrict__ agg,
                        int E) {
    const int tid  = blockIdx.x * blockDim.x + threadIdx.x;
    const int edge = tid >> 5;
    const int lane = tid & 31;
    if (edge >= E) return;
    const int s = __ldg(src + edge);
    const int d = __ldg(dst + edge);
    const float4 v =
        reinterpret_cast<const float4*>(h + (size_t)s * D)[lane];
    float* out = agg + (size_t)d * D + lane * 4;
    unsafeAtomicAdd(out + 0, v.x);
    unsafeAtomicAdd(out + 1, v.y);
    unsafeAtomicAdd(out + 2, v.z);
    unsafeAtomicAdd(out + 3, v.w);
}

// ---------------------------------------------------------------------------
// Fused GIN layer: rst = h + agg; h1 = act(rst@W1 + b1); out = act(h1@W2 + b2)
// One block = 16 node rows x 128 cols. 8 waves; wave w owns output columns
// [16w, 16w+16). Both weight matrices live in LDS (128KB). fp32 WMMA
// (v_wmma_f32_16x16x4_f32), K accumulated over 32 steps of 4.
// ---------------------------------------------------------------------------
__global__ __launch_bounds__(256, 2)
void gin_layer_kernel(const float* __restrict__ hin,
                      const float* __restrict__ agg,
                      const float* __restrict__ W1,
                      const float* __restrict__ b1,
                      const float* __restrict__ W2,
                      const float* __restrict__ b2,
                      float*       __restrict__ hout,
                      int N, int do_relu) {
    __shared__ float sW1[D * D];          // 64 KB
    __shared__ float sW2[D * D];          // 64 KB
    __shared__ float sT [16 * TSTRIDE];   // rst tile, padded
    __shared__ float sH1[16 * TSTRIDE];   // hidden tile, padded

    const int t    = threadIdx.x;         // 0..255
    const int wave = t >> 5;              // 0..7
    const int lane = t & 31;
    const int half = lane >> 4;           // 0: K pair 0/1, 1: K pair 2/3
    const int lm   = lane & 15;
    const int row0 = blockIdx.x * 16;
    const int col0 = wave * 16;

    // --- stage W1/W2 into LDS: 16384 floats each, 16 float4 per thread ---
    {
        const float4* w1v = reinterpret_cast<const float4*>(W1);
        const float4* w2v = reinterpret_cast<const float4*>(W2);
        float4* s1v = reinterpret_cast<float4*>(sW1);
        float4* s2v = reinterpret_cast<float4*>(sW2);
        #pragma unroll
        for (int i = 0; i < (D * D / 4) / 256; ++i) {
            const int idx = t + i * 256;
            s1v[idx] = w1v[idx];
            s2v[idx] = w2v[idx];
        }
    }

    // --- stage rst = hin + agg tile (16 x 128) into padded LDS ---
    for (int i = t; i < 16 * (D / 4); i += 256) {
        const int r  = i / (D / 4);
        const int c4 = i % (D / 4);
        float4 a = make_float4(0.f, 0.f, 0.f, 0.f);
        if (row0 + r < N) {
            const float4 hv =
                reinterpret_cast<const float4*>(hin + (size_t)(row0 + r) * D)[c4];
            const float4 gv =
                reinterpret_cast<const float4*>(agg + (size_t)(row0 + r) * D)[c4];
            a = make_float4(hv.x + gv.x, hv.y + gv.y, hv.z + gv.z, hv.w + gv.w);
        }
        float* p = &sT[r * TSTRIDE + c4 * 4];
        p[0] = a.x; p[1] = a.y; p[2] = a.z; p[3] = a.w;
    }
    __syncthreads();

    // --- GEMM1: 16x16 tile over K=128 via 32 x v_wmma_f32_16x16x4_f32 ---
    v8f acc = {0.f, 0.f, 0.f, 0.f, 0.f, 0.f, 0.f, 0.f};
    #pragma unroll
    for (int k = 0; k < D; k += 4) {
        v2f a, b;
        // A 16x4: lane lm holds row M=lm, K = k+2*half .. k+2*half+1
        a.x = sT[lm * TSTRIDE + k + 2 * half + 0];
        a.y = sT[lm * TSTRIDE + k + 2 * half + 1];
        // B 4x16: lane lm holds col N=col0+lm, same K pair
        b.x = sW1[(k + 2 * half + 0) * D + col0 + lm];
        b.y = sW1[(k + 2 * half + 1) * D + col0 + lm];
        acc = __builtin_amdgcn_wmma_f32_16x16x4_f32(
            false, a, false, b, (short)0, acc, false, false);
    }

    // --- bias + optional relu -> hidden tile in LDS ---
    {
        const float bias = b1[col0 + lm];
        #pragma unroll
        for (int r = 0; r < 8; ++r) {
            float v = acc[r] + bias;
            if (do_relu) v = fmaxf(v, 0.0f);
            sH1[(r + 8 * half) * TSTRIDE + col0 + lm] = v;
        }
    }
    __syncthreads();

    // --- GEMM2 from hidden tile ---
    v8f acc2 = {0.f, 0.f, 0.f, 0.f, 0.f, 0.f, 0.f, 0.f};
    #pragma unroll
    for (int k = 0; k < D; k += 4) {
        v2f a, b;
        a.x = sH1[lm * TSTRIDE + k + 2 * half + 0];
        a.y = sH1[lm * TSTRIDE + k + 2 * half + 1];
        b.x = sW2[(k + 2 * half + 0) * D + col0 + lm];
        b.y = sW2[(k + 2 * half + 1) * D + col0 + lm];
        acc2 = __builtin_amdgcn_wmma_f32_16x16x4_f32(
            false, a, false, b, (short)0, acc2, false, false);
    }

    // --- bias + optional relu -> global output rows ---
    {
        const float bias = b2[col0 + lm];
        #pragma unroll
        for (int r = 0; r < 8; ++r) {
            float v = acc2[r] + bias;
            if (do_relu) v = fmaxf(v, 0.0f);
            const int row = row0 + r + 8 * half;
            if (row < N) hout[(size_t)row * D + col0 + lm] = v;
        }
    }
}

// ---------------------------------------------------------------------------
// Host-side launch: 3 layers of { memset agg, scatter-add, fused layer }.
// ws layout: [ agg : N*D floats ][ hbuf : N*D floats ]
// ---------------------------------------------------------------------------
extern "C" void kernel_launch(void* const* d_in, const int* in_sizes, int n_in,
                              void* d_out, int out_size, void* d_ws, size_t ws_size,
                              hipStream_t stream) {
    const float* feat = (const float*)d_in[0];
    const float* W1   = (const float*)d_in[1];
    const float* b1   = (const float*)d_in[2];
    const float* W2   = (const float*)d_in[3];
    const float* b2   = (const float*)d_in[4];
    const int*   src  = (const int*)d_in[5];
    const int*   dst  = (const int*)d_in[6];
    float*       out  = (float*)d_out;

    const int N = in_sizes[0] / D;
    const int E = in_sizes[5];
    const int L = in_sizes[1] / (D * D);

    float* agg  = (float*)d_ws;
    float* hbuf = agg + (size_t)N * D;

    const int scatter_threads = 256;
    const long long scatter_total = (long long)E * 32;
    const int scatter_grid = (int)((scatter_total + scatter_threads - 1) / scatter_threads);
    const int layer_grid = (N + 15) / 16;

    const float* hcur = feat;
    for (int i = 0; i < L; ++i) {
        hipMemsetAsync(agg, 0, (size_t)N * D * sizeof(float), stream);
        gin_scatter_kernel<<<scatter_grid, scatter_threads, 0, stream>>>(
            hcur, src, dst, agg, E);
        float* hnext = (i == L - 1) ? out : hbuf;
        const int relu = (i < L - 1) ? 1 : 0;
        gin_layer_kernel<<<layer_grid, 256, 0, stream>>>(
            hcur, agg,
            W1 + (size_t)i * D * D, b1 + (size_t)i * D,
            W2 + (size_t)i * D * D, b2 + (size_t)i * D,
            hnext, N, relu);
        hcur = hnext;   // in-place update is safe: each block only touches its own rows
    }
}